// AdverNCE_39994735460891
// MI455X (gfx1250) — compile-verified
//
#include <hip/hip_runtime.h>

// Problem constants (match the reference)
#define B_    256
#define V_    100000
#define K_    1000
#define KP1_  1001
#define EPS_  1e-6f

// Streaming geometry for the top-K kernel: 512 threads = 16 waves,
// each wave stages 32 lanes * 4 floats = 128 floats (512 B) per iteration
// per array via global_load_async_to_lds_b128 (one B128 per lane).
#define NW   16
#define WF   128
#define G_   ((V_ + WF - 1) / WF)      // 782 groups of 128 floats
#define NIT  ((G_ + NW - 1) / NW)      // 49 iterations

// Monotone float -> uint key (larger float => larger key)
__device__ __forceinline__ unsigned mono_key(float f) {
    unsigned b = __float_as_uint(f);
    return b ^ (unsigned)(((int)b >> 31) | (int)0x80000000);
}

__global__ __launch_bounds__(512)
void adver_nce_topk(const float* __restrict__ noise,
                    const float* __restrict__ gum,
                    const int*   __restrict__ target,
                    int*         __restrict__ sel)
{
    __shared__ __align__(16) float s_n[2][NW][WF];   // 16 KB  (double-buffered noise stage)
    __shared__ __align__(16) float s_g[2][NW][WF];   // 16 KB  (double-buffered gumbel stage)
    __shared__ unsigned s_hist[256];
    __shared__ unsigned s_pkey, s_need, s_gt, s_tie;

    const int tid  = threadIdx.x;
    const int w    = tid >> 5;     // wave id (wave32)
    const int lane = tid & 31;
    const int row  = blockIdx.x;

    const float* nrow = noise + (size_t)row * V_;
    const float* grow = gum   + (size_t)row * V_;
    int* selrow = sel + row * 1024;
    const int tgt = target[row];

    // LDS byte addresses for this lane's 16B staging slots (flat addr low 32b = LDS offset)
    unsigned ldsn[2], ldsg[2];
    ldsn[0] = (unsigned)(size_t)&s_n[0][w][lane * 4];
    ldsn[1] = (unsigned)(size_t)&s_n[1][w][lane * 4];
    ldsg[0] = (unsigned)(size_t)&s_g[0][w][lane * 4];
    ldsg[1] = (unsigned)(size_t)&s_g[1][w][lane * 4];

    if (tid < 256) s_hist[tid] = 0u;
    if (tid == 0) { s_pkey = 0u; s_need = K_; s_gt = 0u; s_tie = 0u; }
    __syncthreads();

    // 5 passes over the row: 4 radix-histogram passes (8 bits each) to find the
    // exact K-th key, then 1 selection pass. Each pass streams the row through
    // LDS via async DMA (per-wave-private buffers => no barrier in the loop).
    for (int p = 0; p < 5; ++p) {
        const unsigned pkey  = s_pkey;
        const unsigned needv = s_need;
        const int      shift = (p < 4) ? (24 - 8 * p) : 0;
        const unsigned pmask = (p >= 1 && p < 4) ? (0xFFFFFFFFu << (32 - 8 * p)) : 0u;
        const unsigned T        = pkey;          // pass 4: exact threshold
        const unsigned tie_take = needv;         // pass 4: #keys == T to accept
        const unsigned gt_total = K_ - needv;    // pass 4: #keys  > T

        // Prologue: issue chunk 0 (async DMA into buffer 0)
        {
            const int g0 = w;
            if (g0 < G_) {
                const int e0 = g0 * WF + lane * 4;
                if (e0 < V_) {
                    const unsigned off = (unsigned)e0 * 4u;
                    asm volatile("global_load_async_to_lds_b128 %0, %1, %2"
                                 :: "v"(ldsn[0]), "v"(off), "s"(nrow) : "memory");
                    asm volatile("global_load_async_to_lds_b128 %0, %1, %2"
                                 :: "v"(ldsg[0]), "v"(off), "s"(grow) : "memory");
                }
            }
        }

        for (int it = 0; it < NIT; ++it) {
            // Issue next chunk into the other buffer (wave-uniform decision)
            const int gn = (it + 1) * NW + w;
            const bool overlapped = (it + 1 < NIT) && (gn < G_);
            if (overlapped) {
                const int en   = gn * WF + lane * 4;
                const int nbuf = (it + 1) & 1;
                if (en < V_) {
                    const unsigned off = (unsigned)en * 4u;
                    asm volatile("global_load_async_to_lds_b128 %0, %1, %2"
                                 :: "v"(ldsn[nbuf]), "v"(off), "s"(nrow) : "memory");
                    asm volatile("global_load_async_to_lds_b128 %0, %1, %2"
                                 :: "v"(ldsg[nbuf]), "v"(off), "s"(grow) : "memory");
                }
            }
            // Wait for current chunk: <=2 outstanding leaves the next chunk's
            // two DMAs in flight; on the tail, drain fully.
            if (overlapped) {
                asm volatile("s_wait_asynccnt 0x2" ::: "memory");
            } else {
                asm volatile("s_wait_asynccnt 0x0" ::: "memory");
            }

            const int g = it * NW + w;
            if (g < G_) {
                const int e = g * WF + lane * 4;
                if (e < V_) {
                    const int buf = it & 1;
                    const float4 nv4 = *(const float4*)&s_n[buf][w][lane * 4];
                    const float4 gv4 = *(const float4*)&s_g[buf][w][lane * 4];
                    const float na[4] = {nv4.x, nv4.y, nv4.z, nv4.w};
                    const float ua[4] = {gv4.x, gv4.y, gv4.z, gv4.w};
#pragma unroll
                    for (int k = 0; k < 4; ++k) {
                        const int idx = e + k;
                        // score = logits + Gumbel; log_softmax shift is top-K invariant
                        const float gmb = -__logf(-__logf(ua[k]));
                        unsigned key = mono_key(na[k] + gmb);
                        if (idx == tgt) key = 0u;   // mask the positive (-inf)
                        if (p < 4) {
                            if ((key & pmask) == pkey)
                                atomicAdd(&s_hist[(key >> shift) & 0xFFu], 1u);
                        } else {
                            if (key > T) {
                                const unsigned pos = atomicAdd(&s_gt, 1u);
                                selrow[1 + pos] = idx;
                            } else if (key == T) {
                                const unsigned t = atomicAdd(&s_tie, 1u);
                                if (t < tie_take)
                                    selrow[1 + gt_total + t] = idx;
                            }
                        }
                    }
                }
            }
        }

        __syncthreads();
        if (p < 4) {
            if (tid == 0) {
                unsigned cum = 0;
                int b = 255;
                for (; b > 0; --b) {
                    const unsigned c = s_hist[b];
                    if (cum + c >= needv) break;
                    cum += c;
                }
                s_pkey = pkey | ((unsigned)b << shift);
                s_need = needv - cum;
            }
            __syncthreads();
            if (tid < 256) s_hist[tid] = 0u;
            __syncthreads();
        } else {
            if (tid == 0) selrow[0] = tgt;   // slot 0 = positive item
        }
    }
}

// One block per row: gather 1001 candidates, softmax both logit sets,
// accumulate the row's sum of log-likelihood terms.
__global__ __launch_bounds__(256)
void adver_nce_score(const float* __restrict__ noise,
                     const float* __restrict__ actual,
                     const int*   __restrict__ sel,
                     float*       __restrict__ row_loss)
{
    __shared__ float r1[256];
    __shared__ float r2[256];
    const int tid = threadIdx.x;
    const int row = blockIdx.x;
    const float* nrow = noise  + (size_t)row * V_;
    const float* arow = actual + (size_t)row * V_;
    const int* srow = sel + row * 1024;

    float nv[4], av[4];
    bool ok[4];
    float mn = -3.0e38f, ma = -3.0e38f;
#pragma unroll
    for (int j = 0; j < 4; ++j) {
        const int i = tid + j * 256;
        ok[j] = (i < KP1_);
        if (ok[j]) {
            const int c = srow[i];
            nv[j] = nrow[c];
            av[j] = arow[c];
            mn = fmaxf(mn, nv[j]);
            ma = fmaxf(ma, av[j]);
        } else { nv[j] = 0.f; av[j] = 0.f; }
    }
    r1[tid] = mn; r2[tid] = ma;
    __syncthreads();
    for (int s = 128; s > 0; s >>= 1) {
        if (tid < s) { r1[tid] = fmaxf(r1[tid], r1[tid + s]); r2[tid] = fmaxf(r2[tid], r2[tid + s]); }
        __syncthreads();
    }
    mn = r1[0]; ma = r2[0];
    __syncthreads();

    float en[4], ea[4];
    float sn = 0.f, sa = 0.f;
#pragma unroll
    for (int j = 0; j < 4; ++j) {
        if (ok[j]) {
            en[j] = __expf(nv[j] - mn);
            ea[j] = __expf(av[j] - ma);
            sn += en[j]; sa += ea[j];
        } else { en[j] = 0.f; ea[j] = 0.f; }
    }
    r1[tid] = sn; r2[tid] = sa;
    __syncthreads();
    for (int s = 128; s > 0; s >>= 1) {
        if (tid < s) { r1[tid] += r1[tid + s]; r2[tid] += r2[tid + s]; }
        __syncthreads();
    }
    const float izn = 1.0f / r1[0];
    const float iza = 1.0f / r2[0];
    __syncthreads();

    float ls = 0.f;
#pragma unroll
    for (int j = 0; j < 4; ++j) {
        if (ok[j]) {
            const float np = en[j] * izn;
            const float ap = ea[j] * iza;
            const float deno = (float)K_ * np + ap + EPS_;
            const int i = tid + j * 256;
            const float r = (i == 0) ? ap : np;   // slot 0 uses actual ratio
            ls += __logf(r / deno);
        }
    }
    r1[tid] = ls;
    __syncthreads();
    for (int s = 128; s > 0; s >>= 1) {
        if (tid < s) r1[tid] += r1[tid + s];
        __syncthreads();
    }
    if (tid == 0) row_loss[row] = r1[0];
}

__global__ __launch_bounds__(256)
void adver_nce_final(const float* __restrict__ row_loss, float* __restrict__ out)
{
    __shared__ float r[256];
    const int tid = threadIdx.x;
    r[tid] = row_loss[tid];
    __syncthreads();
    for (int s = 128; s > 0; s >>= 1) {
        if (tid < s) r[tid] += r[tid + s];
        __syncthreads();
    }
    if (tid == 0) out[0] = -r[0] / (float)(B_ * KP1_);
}

extern "C" void kernel_launch(void* const* d_in, const int* in_sizes, int n_in,
                              void* d_out, int out_size, void* d_ws, size_t ws_size,
                              hipStream_t stream)
{
    const float* noise  = (const float*)d_in[0];  // noise_logits   [B,V] f32
    const float* actual = (const float*)d_in[1];  // actual_full    [B,V] f32
    const int*   target = (const int*)  d_in[2];  // target_id      [B]   i32
    const float* gum    = (const float*)d_in[3];  // gumbel_u       [B,V] f32

    int*   sel      = (int*)d_ws;                                      // [B,1024] indices
    float* row_loss = (float*)((char*)d_ws + (size_t)B_ * 1024 * 4);   // [B] partial losses

    adver_nce_topk <<<B_, 512, 0, stream>>>(noise, gum, target, sel);
    adver_nce_score<<<B_, 256, 0, stream>>>(noise, actual, sel, row_loss);
    adver_nce_final<<<1,  256, 0, stream>>>(row_loss, (float*)d_out);

    (void)in_sizes; (void)n_in; (void)out_size; (void)ws_size;
}